// ImprovedMethaneGNN_77326591197790
// MI455X (gfx1250) — compile-verified
//
#include <hip/hip_runtime.h>
#include <math.h>

#define N_NODES 50000
#define N_EDGES 640000
#define HDIM    128
#define NGRAPH  256
#define BN_EPS  1e-5f

typedef float v2f __attribute__((ext_vector_type(2)));
typedef float v8f __attribute__((ext_vector_type(8)));

// ---------------------------------------------------------------------------
// fp32 GEMM: H[N,128] = X[N,128] @ W[128,128] via V_WMMA_F32_16X16X4_F32.
// Block = 256 threads = 8 waves; block owns a 16-row M-tile (staged in LDS),
// wave w owns N-tile [16w, 16w+16). K-loop of 32 WMMA ops per wave.
// A layout (16x4 f32): lanes 0-15 -> M=lane, VGPR0/1 = K,K+1; lanes 16-31 add K+2.
// B layout (4x16):     VGPR r, lanes 0-15 -> K=r, N=lane; lanes 16-31 -> K=r+2.
// C/D layout (16x16):  VGPR r, lanes 0-15 -> M=r; lanes 16-31 -> M=r+8; N=lane&15.
// ---------------------------------------------------------------------------
__global__ __launch_bounds__(256) void gemm_xw(const float* __restrict__ X,
                                               const float* __restrict__ W,
                                               float* __restrict__ H) {
  __shared__ float sX[16 * 128];
  const int m0  = blockIdx.x * 16;
  const int tid = threadIdx.x;
#pragma unroll
  for (int i = 0; i < 8; ++i)
    sX[tid + 256 * i] = X[(size_t)m0 * 128 + tid + 256 * i];
  __syncthreads();

  const int wave = tid >> 5;
  const int lane = tid & 31;
  const int n0   = wave * 16;
  const int half = lane >> 4;   // 0 or 1
  const int l16  = lane & 15;

  v8f acc = {};
#pragma unroll 4
  for (int k = 0; k < 128; k += 4) {
    const int kk = k + 2 * half;
    v2f a, b;
    a.x = sX[l16 * 128 + kk];
    a.y = sX[l16 * 128 + kk + 1];
    b.x = W[kk * 128 + n0 + l16];
    b.y = W[(kk + 1) * 128 + n0 + l16];
    acc = __builtin_amdgcn_wmma_f32_16x16x4_f32(false, a, false, b,
                                                (short)0, acc, false, false);
  }
#pragma unroll
  for (int r = 0; r < 8; ++r)
    H[(size_t)(m0 + r + 8 * half) * 128 + n0 + l16] = acc[r];
}

// ---------------------------------------------------------------------------
// Per (node, head): l_src = <h, a_src>, l_dst = <h, a_dst>
// ---------------------------------------------------------------------------
__global__ void attn_logits(const float* __restrict__ H,
                            const float* __restrict__ a_src,
                            const float* __restrict__ a_dst,
                            float* __restrict__ lsrc, float* __restrict__ ldst,
                            int heads, int dh) {
  int t = blockIdx.x * blockDim.x + threadIdx.x;
  if (t >= N_NODES * heads) return;
  int n = t / heads, h = t - n * heads;
  const float* hp = H + (size_t)n * HDIM + h * dh;
  const float* as = a_src + h * dh;
  const float* ad = a_dst + h * dh;
  float s0 = 0.f, s1 = 0.f;
  for (int d = 0; d < dh; ++d) { float v = hp[d]; s0 += v * as[d]; s1 += v * ad[d]; }
  lsrc[t] = s0; ldst[t] = s1;
}

__global__ void layer_init(float* __restrict__ acc, float* __restrict__ nmax,
                           float* __restrict__ nsum, int heads) {
  int t = blockIdx.x * blockDim.x + threadIdx.x;
  if (t < N_NODES * HDIM) acc[t] = 0.f;
  if (t < N_NODES * heads) { nmax[t] = -__builtin_huge_valf(); nsum[t] = 0.f; }
}

// float atomic-max via ordered-int trick (single GLOBAL_ATOMIC per call, no CAS loop)
__device__ __forceinline__ void atomicMaxF(float* addr, float v) {
  if (v >= 0.f) atomicMax((int*)addr, __float_as_int(v));
  else          atomicMin((unsigned int*)addr, __float_as_uint(v));
}

// Pass 1 over edges (incl. self loops): segment max of leaky_relu logits per dst/head
__global__ void edge_max(const int* __restrict__ ei,
                         const float* __restrict__ lsrc, const float* __restrict__ ldst,
                         float* __restrict__ nmax, int heads, int etot) {
  int t = blockIdx.x * blockDim.x + threadIdx.x;
  if (t >= etot * heads) return;
  int e = t / heads, h = t - e * heads;
  int s, d;
  if (e < N_EDGES) { s = ei[e]; d = ei[N_EDGES + e]; } else { s = d = e - N_EDGES; }
  float l = lsrc[s * heads + h] + ldst[d * heads + h];
  l = l > 0.f ? l : 0.2f * l;
  atomicMaxF(&nmax[d * heads + h], l);
}

// Pass 2: e = exp(logit - max[dst]); nsum[dst] += e; acc[dst,:] += e * h[src,:]
// (normalization by nsum deferred to finalize -> only 2 edge passes total)
__global__ __launch_bounds__(128) void edge_accum(const int* __restrict__ ei,
                            const float* __restrict__ lsrc, const float* __restrict__ ldst,
                            const float* __restrict__ nmax, float* __restrict__ nsum,
                            const float* __restrict__ H, float* __restrict__ acc,
                            int heads, int dh) {
  __shared__ float se[8];
  const int e = blockIdx.x;
  int s, d;
  if (e < N_EDGES) { s = ei[e]; d = ei[N_EDGES + e]; } else { s = d = e - N_EDGES; }
  const int f = threadIdx.x;
  if (f < heads) {
    float l = lsrc[s * heads + f] + ldst[d * heads + f];
    l = l > 0.f ? l : 0.2f * l;
    float ev = expf(l - nmax[d * heads + f]);
    atomicAdd(&nsum[d * heads + f], ev);
    se[f] = ev;
  }
  __syncthreads();
  float ev = se[f / dh];
  atomicAdd(&acc[(size_t)d * HDIM + f], ev * H[(size_t)s * HDIM + f]);
}

// out = relu(BN(acc/nsum + b)) [+ residual]
__global__ void finalize(const float* __restrict__ acc, const float* __restrict__ nsum,
                         const float* __restrict__ bias, const float* __restrict__ g,
                         const float* __restrict__ beta, const float* __restrict__ mu,
                         const float* __restrict__ var, const float* __restrict__ xin,
                         float* __restrict__ xout, int heads, int dh, int add_res) {
  int t = blockIdx.x * blockDim.x + threadIdx.x;
  if (t >= N_NODES * HDIM) return;
  int f = t & 127;
  int n = t >> 7;
  int h = f / dh;
  float v = acc[t] / nsum[n * heads + h] + bias[f];
  v = (v - mu[f]) * g[f] * rsqrtf(var[f] + BN_EPS) + beta[f];
  v = v > 0.f ? v : 0.f;
  if (add_res) v += xin[t];
  xout[t] = v;
}

// ----------------------------- pooling --------------------------------------
__global__ void pool_init(float* wmax, float* ssum, float* cnt,
                          float* xadd, float* xattn) {
  int t = blockIdx.x * blockDim.x + threadIdx.x;
  if (t < NGRAPH) { wmax[t] = -__builtin_huge_valf(); ssum[t] = 0.f; cnt[t] = 0.f; }
  if (t < NGRAPH * HDIM) { xadd[t] = 0.f; xattn[t] = 0.f; }
}

// per-node attention score w = tanh(x@A1 + ab1) @ A2 + ab2 ; segment max over graphs
__global__ __launch_bounds__(64) void node_score(const float* __restrict__ x,
                           const int* __restrict__ batch,
                           const float* __restrict__ A1, const float* __restrict__ ab1,
                           const float* __restrict__ A2, const float* __restrict__ ab2,
                           float* __restrict__ w, float* __restrict__ wmax) {
  __shared__ float red[64];
  const int n = blockIdx.x;
  const int j = threadIdx.x;
  const float* xp = x + (size_t)n * HDIM;
  float t = ab1[j];
  for (int k = 0; k < HDIM; ++k) t += xp[k] * A1[k * 64 + j];
  red[j] = tanhf(t) * A2[j];
  __syncthreads();
  for (int off = 32; off > 0; off >>= 1) {
    if (j < off) red[j] += red[j + off];
    __syncthreads();
  }
  if (j == 0) {
    float wn = red[0] + ab2[0];
    w[n] = wn;
    atomicMaxF(&wmax[batch[n]], wn);
  }
}

// per-node: accumulate cnt, sum(e), sum(x), sum(x*e) into per-graph buffers
__global__ __launch_bounds__(128) void node_pool(const float* __restrict__ x,
                          const int* __restrict__ batch,
                          const float* __restrict__ w, const float* __restrict__ wmax,
                          float* __restrict__ ssum, float* __restrict__ cnt,
                          float* __restrict__ xadd, float* __restrict__ xattn) {
  __shared__ float se;
  const int n = blockIdx.x;
  const int f = threadIdx.x;
  const int g = batch[n];
  if (f == 0) {
    float e = expf(w[n] - wmax[g]);
    se = e;
    atomicAdd(&ssum[g], e);
    atomicAdd(&cnt[g], 1.f);
  }
  __syncthreads();
  float xv = x[(size_t)n * HDIM + f];
  atomicAdd(&xadd[(size_t)g * HDIM + f], xv);
  atomicAdd(&xattn[(size_t)g * HDIM + f], xv * se);
}

// per-graph: c = mean + 0.5*attn + 0.1*sum ; out = M3ᵀ relu(M2ᵀ relu(M1ᵀ c + b1)+b2)+b3
__global__ __launch_bounds__(128) void readout(const float* __restrict__ xadd,
                        const float* __restrict__ xattn,
                        const float* __restrict__ ssum, const float* __restrict__ cnt,
                        const float* __restrict__ M1, const float* __restrict__ mb1,
                        const float* __restrict__ M2, const float* __restrict__ mb2,
                        const float* __restrict__ M3, const float* __restrict__ mb3,
                        float* __restrict__ out) {
  __shared__ float c[128], h1[128], h2[64];
  const int g = blockIdx.x;
  const int t = threadIdx.x;
  float cg = fmaxf(cnt[g], 1.f);
  float s  = ssum[g]; s = s > 0.f ? s : 1.f;
  float xa = xadd[(size_t)g * HDIM + t];
  c[t] = xa / cg + 0.5f * (xattn[(size_t)g * HDIM + t] / s) + 0.1f * xa;
  __syncthreads();
  float a = mb1[t];
  for (int k = 0; k < 128; ++k) a += c[k] * M1[k * 128 + t];
  h1[t] = fmaxf(a, 0.f);
  __syncthreads();
  if (t < 64) {
    float b = mb2[t];
    for (int k = 0; k < 128; ++k) b += h1[k] * M2[k * 64 + t];
    h2[t] = fmaxf(b, 0.f) * M3[t];
  }
  __syncthreads();
  for (int off = 32; off > 0; off >>= 1) {
    if (t < off) h2[t] += h2[t + off];
    __syncthreads();
  }
  if (t == 0) out[g] = h2[0] + mb3[0];
}

// ---------------------------------------------------------------------------
extern "C" void kernel_launch(void* const* d_in, const int* in_sizes, int n_in,
                              void* d_out, int out_size, void* d_ws, size_t ws_size,
                              hipStream_t stream) {
  (void)in_sizes; (void)n_in; (void)out_size; (void)ws_size;
  const float* x0    = (const float*)d_in[0];
  const int*   ei    = (const int*)d_in[1];   // [2,E]: src then dst
  const int*   batch = (const int*)d_in[3];
  // params: per layer i -> d_in[4+8i+{0..7}] = W, as, ad, b, bn_g, bn_b, bn_m, bn_v
  const float* A1  = (const float*)d_in[36]; const float* ab1 = (const float*)d_in[37];
  const float* A2  = (const float*)d_in[38]; const float* ab2 = (const float*)d_in[39];
  const float* M1  = (const float*)d_in[40]; const float* mb1 = (const float*)d_in[41];
  const float* M2  = (const float*)d_in[42]; const float* mb2 = (const float*)d_in[43];
  const float* M3  = (const float*)d_in[44]; const float* mb3 = (const float*)d_in[45];

  float* ws = (float*)d_ws;
  size_t off = 0;
  float* xA    = ws + off; off += (size_t)N_NODES * HDIM;
  float* xB    = ws + off; off += (size_t)N_NODES * HDIM;
  float* Hbuf  = ws + off; off += (size_t)N_NODES * HDIM;
  float* accb  = ws + off; off += (size_t)N_NODES * HDIM;
  float* lsrc  = ws + off; off += (size_t)N_NODES * 8;
  float* ldst  = ws + off; off += (size_t)N_NODES * 8;
  float* nmax  = ws + off; off += (size_t)N_NODES * 8;
  float* nsum  = ws + off; off += (size_t)N_NODES * 8;
  float* wbuf  = ws + off; off += (size_t)N_NODES;
  float* wmax  = ws + off; off += NGRAPH;
  float* ssum  = ws + off; off += NGRAPH;
  float* cntb  = ws + off; off += NGRAPH;
  float* xadd  = ws + off; off += (size_t)NGRAPH * HDIM;
  float* xattn = ws + off; off += (size_t)NGRAPH * HDIM;

  const int HEADS[4] = {8, 8, 8, 1};
  const int etot = N_EDGES + N_NODES;   // with self loops

  const float* xcur = x0;
  float* xnext = xA;
  for (int i = 0; i < 4; ++i) {
    const float* W    = (const float*)d_in[4 + i * 8 + 0];
    const float* as_  = (const float*)d_in[4 + i * 8 + 1];
    const float* ad_  = (const float*)d_in[4 + i * 8 + 2];
    const float* b_   = (const float*)d_in[4 + i * 8 + 3];
    const float* g_   = (const float*)d_in[4 + i * 8 + 4];
    const float* be_  = (const float*)d_in[4 + i * 8 + 5];
    const float* mu_  = (const float*)d_in[4 + i * 8 + 6];
    const float* var_ = (const float*)d_in[4 + i * 8 + 7];
    const int heads = HEADS[i];
    const int dh = HDIM / heads;

    gemm_xw<<<N_NODES / 16, 256, 0, stream>>>(xcur, W, Hbuf);
    attn_logits<<<(N_NODES * heads + 255) / 256, 256, 0, stream>>>(
        Hbuf, as_, ad_, lsrc, ldst, heads, dh);
    layer_init<<<(N_NODES * HDIM + 255) / 256, 256, 0, stream>>>(accb, nmax, nsum, heads);
    edge_max<<<(etot * heads + 255) / 256, 256, 0, stream>>>(ei, lsrc, ldst, nmax, heads, etot);
    edge_accum<<<etot, 128, 0, stream>>>(ei, lsrc, ldst, nmax, nsum, Hbuf, accb, heads, dh);
    finalize<<<(N_NODES * HDIM + 255) / 256, 256, 0, stream>>>(
        accb, nsum, b_, g_, be_, mu_, var_, xcur, xnext, heads, dh, i > 0);

    xcur = xnext;
    xnext = (xcur == xA) ? xB : xA;
  }

  pool_init<<<(NGRAPH * HDIM + 255) / 256, 256, 0, stream>>>(wmax, ssum, cntb, xadd, xattn);
  node_score<<<N_NODES, 64, 0, stream>>>(xcur, batch, A1, ab1, A2, ab2, wbuf, wmax);
  node_pool<<<N_NODES, 128, 0, stream>>>(xcur, batch, wbuf, wmax, ssum, cntb, xadd, xattn);
  readout<<<NGRAPH, 128, 0, stream>>>(xadd, xattn, ssum, cntb,
                                      M1, mb1, M2, mb2, M3, mb3, (float*)d_out);
}